// BEVFeature_61735859912946
// MI455X (gfx1250) — compile-verified
//
#include <hip/hip_runtime.h>
#include <cstddef>

// ---------------- constants ----------------
#define EMBED   128
#define HEADS   8
#define LEVELS  3
#define POINTS  4
#define HDIM    16
#define SX      128
#define SY      128
#define SZ      16
#define L_TOT   (SX*SY*SZ)          // 262144
#define IMG_H   192
#define IMG_W   640
#define NPIX    (IMG_H*IMG_W)       // 122880
#define NUM_CLS 20
// value maps (rows = Hl*Wl per level, concatenated)
#define V0_H 24
#define V0_W 80
#define V1_H 12
#define V1_W 40
#define V2_H 6
#define V2_W 20
#define VBASE0 0
#define VBASE1 (V0_H*V0_W)            // 1920
#define VBASE2 (VBASE1 + V1_H*V1_W)   // 2400
#define VROWS  (VBASE2 + V2_H*V2_W)   // 2520

typedef __attribute__((ext_vector_type(2))) float v2f;
typedef __attribute__((ext_vector_type(8))) float v8f;

__device__ inline v8f vzero8() {
  v8f z;
#pragma unroll
  for (int i = 0; i < 8; ++i) z[i] = 0.f;
  return z;
}

// CDNA5 fp32 WMMA: D(16x16) = A(16x4) * B(4x16) + C, wave32.
// A frag: lane l -> m = l&15, k0 = (l>>4)*2, {A[m][k0], A[m][k0+1]}
// B frag: lane l -> n = l&15, k0 = (l>>4)*2, {B[k0][n], B[k0+1][n]}
// C/D   : lane l -> n = l&15, rows (l>>4)*8 + r in regs r=0..7
__device__ inline v8f wmma_f32(v2f a, v2f b, v8f c) {
  return __builtin_amdgcn_wmma_f32_16x16x4_f32(false, a, false, b, (short)0, c, false, false);
}

// ---------------- B-operand fragment packing ----------------
// W is (K x N) row-major. P[tile][ks][lane][2] with
//   k   = ks*4 + (lane>>4)*2 + j
//   col = tile*16 + (lane&15)       (zero-padded past N)
// so a lane's B fragment is ONE contiguous 8-byte load, coalesced across the wave.
__global__ void packB_kernel(const float* __restrict__ W, float* __restrict__ P,
                             int K, int N, int Ntiles) {
  int total = Ntiles * (K / 4) * 64;
  int t = blockIdx.x * blockDim.x + threadIdx.x;
  if (t >= total) return;
  int j    = t & 1;
  int lane = (t >> 1) & 31;
  int ks   = (t >> 6) % (K / 4);
  int tile = (t >> 6) / (K / 4);
  int k    = ks*4 + (lane >> 4)*2 + j;
  int col  = tile*16 + (lane & 15);
  P[t] = (col < N) ? W[k*N + col] : 0.f;
}

// ---------------- small prep: matrix inverses ----------------
__device__ inline void inv3(const float* m, float* o) {
  float a=m[0],b=m[1],c=m[2],d=m[3],e=m[4],f=m[5],g=m[6],h=m[7],i=m[8];
  float A =  (e*i - f*h), B = -(d*i - f*g), C =  (d*h - e*g);
  float det = a*A + b*B + c*C;
  float id = 1.f/det;
  o[0]=A*id;            o[1]=-(b*i - c*h)*id; o[2]= (b*f - c*e)*id;
  o[3]=B*id;            o[4]= (a*i - c*g)*id; o[5]=-(a*f - c*d)*id;
  o[6]=C*id;            o[7]=-(a*h - b*g)*id; o[8]= (a*e - b*d)*id;
}

// ws_inv layout: [0..8] invK, [9..17] invR(E), [18..20] -invR*t, [21..24] inv(post_rot 2x2)
__global__ void prep_kernel(const float* __restrict__ K, const float* __restrict__ E,
                            const float* __restrict__ post_rot, float* __restrict__ ws) {
  if (threadIdx.x != 0 || blockIdx.x != 0) return;
  inv3(K, ws);
  float R[9];
#pragma unroll
  for (int r = 0; r < 3; ++r)
    for (int c = 0; c < 3; ++c) R[r*3+c] = E[r*4+c];
  float Ri[9]; inv3(R, Ri);
#pragma unroll
  for (int i = 0; i < 9; ++i) ws[9+i] = Ri[i];
  float tx = E[3], ty = E[7], tz = E[11];
  for (int r = 0; r < 3; ++r) ws[18+r] = -(Ri[r*3+0]*tx + Ri[r*3+1]*ty + Ri[r*3+2]*tz);
  float a = post_rot[0], b = post_rot[1], c = post_rot[3], d = post_rot[4];
  float id = 1.f/(a*d - b*c);
  ws[21]= d*id; ws[22]=-b*id; ws[23]=-c*id; ws[24]= a*id;
}

// ---------------- positional-encoding projection ----------------
__global__ void posproj_kernel(const float* __restrict__ src, const float* __restrict__ W,
                               const float* __restrict__ bias, float* __restrict__ out, int rows) {
  int t = blockIdx.x * blockDim.x + threadIdx.x;
  if (t >= rows * 512) return;
  int r = t >> 9, j = t & 511;
  float s = bias ? bias[j] : 0.f;
  for (int c = 0; c < EMBED; ++c) s = fmaf(src[r*EMBED + c], W[c*512 + j], s);
  out[t] = s;
}

// ---------------- value projection for all 3 levels ----------------
__global__ void valproj_kernel(const float* __restrict__ f0, const float* __restrict__ f1,
                               const float* __restrict__ f2, const float* __restrict__ w_val,
                               const float* __restrict__ b_val, float* __restrict__ vmap) {
  int t = blockIdx.x * blockDim.x + threadIdx.x;
  if (t >= VROWS * EMBED) return;
  int row = t >> 7, o = t & 127;
  const float* f; int HW, loc;
  if (row < VBASE1)      { f = f0; HW = V0_H*V0_W; loc = row; }
  else if (row < VBASE2) { f = f1; HW = V1_H*V1_W; loc = row - VBASE1; }
  else                   { f = f2; HW = V2_H*V2_W; loc = row - VBASE2; }
  float s = b_val[o];
  for (int c = 0; c < EMBED; ++c) s = fmaf(f[c*HW + loc], w_val[c*EMBED + o], s);
  vmap[t] = s;
}

// ---------------- FOV mask ----------------
__global__ void maskzero_kernel(unsigned* __restrict__ mask) {
  int t = blockIdx.x * blockDim.x + threadIdx.x;
  if (t < L_TOT) mask[t] = 0u;
}

__global__ void maskscatter_kernel(const float* __restrict__ depth, const float* __restrict__ vox_origin,
                                   const float* __restrict__ post_tran, const float* __restrict__ wsinv,
                                   unsigned* __restrict__ mask) {
  int t = blockIdx.x * blockDim.x + threadIdx.x;
  if (t >= NPIX) return;
  int gx = t % IMG_W, gy = t / IMG_W;
  float px = (float)gx - post_tran[0];
  float py = (float)gy - post_tran[1];
  float ux = wsinv[21]*px + wsinv[22]*py;
  float uy = wsinv[23]*px + wsinv[24]*py;
  float d = depth[t];
  float hx = ux*d, hy = uy*d, hz = d;
  float cx = wsinv[0]*hx + wsinv[1]*hy + wsinv[2]*hz;
  float cy = wsinv[3]*hx + wsinv[4]*hy + wsinv[5]*hz;
  float cz = wsinv[6]*hx + wsinv[7]*hy + wsinv[8]*hz;
  float wx = wsinv[ 9]*cx + wsinv[10]*cy + wsinv[11]*cz + wsinv[18];
  float wy = wsinv[12]*cx + wsinv[13]*cy + wsinv[14]*cz + wsinv[19];
  float wz = wsinv[15]*cx + wsinv[16]*cy + wsinv[17]*cz + wsinv[20];
  int ix = (int)((wx - vox_origin[0]) / 0.2f);   // trunc toward zero == jnp.trunc->int32
  int iy = (int)((wy - vox_origin[1]) / 0.2f);
  int iz = (int)((wz - vox_origin[2]) / 0.2f);
  if (ix >= 0 && ix < SX && iy >= 0 && iy < SY && iz >= 0 && iz < SZ)
    mask[ix*(SY*SZ) + iy*SZ + iz] = 1u;   // benign race: all writers store 1
}

// ---------------- deformable attention (fused, WMMA) ----------------
// Block = 64 threads = 2 waves; each wave owns 16 queries.
// LDS: regionA = qv tile 16x128 (reused as headout 16x128)
//      regionB = off/logits 16x288 (reused as out-proj result 16x128)
#define CWAVES 2
__global__ __launch_bounds__(64) void attn_kernel(
    const float* __restrict__ volume,     // (128, L) channel-major
    const float* __restrict__ proj_pix,   // (L, 2)
    const float* __restrict__ px_proj, const float* __restrict__ py_proj,
    const float* __restrict__ pz_proj,    // bias folded into pz
    const v2f* __restrict__ woff_p,       // packed (12 tiles)
    const v2f* __restrict__ wattn_p,      // packed (6 tiles)
    const v2f* __restrict__ wout_p,       // packed (8 tiles)
    const float* __restrict__ b_off, const float* __restrict__ b_attn,
    const float* __restrict__ vmap,       // (2520, 128)
    const float* __restrict__ b_out,
    const float* __restrict__ ln_g, const float* __restrict__ ln_b,
    const unsigned* __restrict__ mask,
    float* __restrict__ vol_out) {
  __shared__ float smem[CWAVES*16*128 + CWAVES*16*288];  // 53 KB
  const int wave = threadIdx.x >> 5;
  const int lane = threadIdx.x & 31;
  const int qbase = blockIdx.x * (CWAVES*16) + wave*16;
  float* regA = smem + wave*(16*128);
  float* regB = smem + CWAVES*(16*128) + wave*(16*288);

  // ---- phase 1: qv = scene_embed + scene_pos into LDS ----
  {
    const int m = lane & 15;
    const int q = qbase + m;
    // pos layout: q = a*2048 + sx*1024 + b*8 + sy*4 + zq
    const int a  = q >> 11;
    const int sx = (q >> 10) & 1;
    const int b  = (q >> 3) & 127;
    const int sy = (q >> 2) & 1;
    const int zq = q & 3;
    const int blk = (sx*2 + sy) * EMBED;
    const float* px = px_proj + a*512 + blk;
    const float* py = py_proj + b*512 + blk;
    const float* pz = pz_proj + zq*512 + blk;
    const int chalf = lane >> 4;
    for (int j = 0; j < 64; ++j) {
      int c = 2*j + chalf;
      float e = volume[(size_t)c * L_TOT + q];
      regA[m*128 + c] = e + px[c] + py[c] + pz[c];
    }
  }
  __syncthreads();

  // ---- phase 2: [off|logits](16x288) = qv(16x128) @ [w_off|w_attn] ----
  {
    const int mA   = lane & 15;
    const int koff = (lane >> 4) * 2;
    const int mb   = (lane >> 4) * 8;
    // offsets: 12 tiles of w_off
    for (int t = 0; t < 12; ++t) {
      const int col = t*16 + (lane & 15);
      const v2f* bp = woff_p + (size_t)t*1024 + lane;   // [ks*32] stride
      v8f acc = vzero8();
#pragma unroll 4
      for (int ks = 0; ks < 32; ++ks) {
        const int k0 = ks*4 + koff;
        v2f a; a.x = regA[mA*128 + k0]; a.y = regA[mA*128 + k0 + 1];
        acc = wmma_f32(a, bp[ks*32], acc);
      }
      const float bias = b_off[col];
#pragma unroll
      for (int r = 0; r < 8; ++r) regB[(mb + r)*288 + col] = acc[r] + bias;
    }
    // attention logits: 6 tiles of w_attn
    for (int t = 0; t < 6; ++t) {
      const int col = 192 + t*16 + (lane & 15);
      const v2f* bp = wattn_p + (size_t)t*1024 + lane;
      v8f acc = vzero8();
#pragma unroll 4
      for (int ks = 0; ks < 32; ++ks) {
        const int k0 = ks*4 + koff;
        v2f a; a.x = regA[mA*128 + k0]; a.y = regA[mA*128 + k0 + 1];
        acc = wmma_f32(a, bp[ks*32], acc);
      }
      const float bias = b_attn[col - 192];
#pragma unroll
      for (int r = 0; r < 8; ++r) regB[(mb + r)*288 + col] = acc[r] + bias;
    }
  }
  __syncthreads();

  // ---- phase 3: softmax + bilinear sampling; headout -> regA ----
  {
    const int lvlW[3] = {V0_W, V1_W, V2_W};
    const int lvlH[3] = {V0_H, V1_H, V2_H};
    const int lvlB[3] = {VBASE0, VBASE1, VBASE2};
    for (int it = 0; it < 4; ++it) {
      const int task = lane + 32*it;       // 16 queries x 8 heads
      const int m = task >> 3, h = task & 7;
      const float* lg = regB + m*288 + 192 + h*12;
      float mx = -1e30f;
#pragma unroll
      for (int j = 0; j < 12; ++j) mx = fmaxf(mx, lg[j]);
      float aw[12]; float s = 0.f;
#pragma unroll
      for (int j = 0; j < 12; ++j) { aw[j] = expf(lg[j] - mx); s += aw[j]; }
      const float sinv = 1.f / s;
      const int q = qbase + m;
      const float refx = (proj_pix[q*2 + 0] + 0.5f) * (1.0f/IMG_W);
      const float refy = (proj_pix[q*2 + 1] + 0.5f) * (1.0f/IMG_H);
      float out[HDIM];
#pragma unroll
      for (int d = 0; d < HDIM; ++d) out[d] = 0.f;
      const float* offp = regB + m*288 + h*24;   // (h*3+l)*8 + p*2
      for (int l3 = 0; l3 < 3; ++l3) {
        const int Wl = lvlW[l3], Hl = lvlH[l3], base = lvlB[l3];
        for (int p = 0; p < 4; ++p) {
          const float ox = offp[l3*8 + p*2], oy = offp[l3*8 + p*2 + 1];
          const float x = refx*(float)Wl + ox - 0.5f;
          const float y = refy*(float)Hl + oy - 0.5f;
          const float xf = floorf(x), yf = floorf(y);
          const float wx = x - xf, wy = y - yf;
          const int x0 = (int)xf, y0 = (int)yf;
          const float a_ = aw[l3*4 + p] * sinv;
          const float cw[4] = {(1.f-wx)*(1.f-wy)*a_, wx*(1.f-wy)*a_,
                               (1.f-wx)*wy*a_,       wx*wy*a_};
          const int cx[4] = {x0, x0+1, x0,   x0+1};
          const int cy[4] = {y0, y0,   y0+1, y0+1};
#pragma unroll
          for (int cr = 0; cr < 4; ++cr) {
            const int xi = cx[cr], yi = cy[cr];
            if (xi < 0 || xi >= Wl || yi < 0 || yi >= Hl) continue;  // zero padding
            const float* vp = vmap + (size_t)(base + yi*Wl + xi)*EMBED + h*HDIM;
            const float w = cw[cr];
#pragma unroll
            for (int d = 0; d < HDIM; ++d) out[d] = fmaf(w, vp[d], out[d]);
          }
        }
      }
#pragma unroll
      for (int d = 0; d < HDIM; ++d) regA[m*128 + h*HDIM + d] = out[d];
    }
  }
  __syncthreads();

  // ---- phase 4: res(16x128) = headout @ w_out + b_out + identity -> regB ----
  {
    const int mA   = lane & 15;
    const int koff = (lane >> 4) * 2;
    const int mb   = (lane >> 4) * 8;
    for (int t = 0; t < 8; ++t) {
      const int col = t*16 + (lane & 15);
      const v2f* bp = wout_p + (size_t)t*1024 + lane;
      v8f acc = vzero8();
#pragma unroll 4
      for (int ks = 0; ks < 32; ++ks) {
        const int k0 = ks*4 + koff;
        v2f a; a.x = regA[mA*128 + k0]; a.y = regA[mA*128 + k0 + 1];
        acc = wmma_f32(a, bp[ks*32], acc);
      }
#pragma unroll
      for (int r = 0; r < 8; ++r) {
        const int m2 = mb + r;
        const float e = volume[(size_t)col * L_TOT + (qbase + m2)];
        regB[m2*128 + col] = acc[r] + b_out[col] + e;
      }
    }
  }
  __syncthreads();

  // ---- phase 5: LayerNorm + masked select + store ----
  {
    const int m = lane & 15;
    const int half = lane >> 4;
    const float* row = regB + m*128;
    float s1 = 0.f, s2 = 0.f;
    for (int j = 0; j < 64; ++j) { float v = row[half*64 + j]; s1 += v; s2 += v*v; }
    s1 += __shfl_xor(s1, 16, 32);
    s2 += __shfl_xor(s2, 16, 32);
    const float mean = s1 * (1.f/128.f);
    const float var  = s2 * (1.f/128.f) - mean*mean;
    const float rstd = rsqrtf(var + 1e-5f);
    const int q = qbase + m;
    const unsigned mk = mask[q];
    for (int j = 0; j < 64; ++j) {
      const int c = half*64 + j;
      const float v = row[c];
      const float y = (v - mean)*rstd*ln_g[c] + ln_b[c];
      const float e = volume[(size_t)c * L_TOT + q];
      vol_out[(size_t)c * L_TOT + q] = mk ? y : e;
    }
  }
}

// ---------------- BEV pool + conv1(relu) + conv2 + seg (WMMA) ----------------
// Block = 128 threads = 4 waves; each wave owns 16 BEV pixels.
#define BWAVES 4
__global__ __launch_bounds__(128) void bev_kernel(
    const float* __restrict__ vol_out,    // (128, L): c*L + pix*16 + z
    const v2f* __restrict__ conv1_p,      // packed (8 tiles, K=1024)
    const float* __restrict__ conv1_b,
    const v2f* __restrict__ conv2_p,      // packed (8 tiles, K=128)
    const float* __restrict__ conv2_b,
    const v2f* __restrict__ seg_p,        // packed (2 tiles, K=128, zero-padded cols)
    const float* __restrict__ seg_b,
    float* __restrict__ bev_f_out, float* __restrict__ seg_out) {
  __shared__ float stage[BWAVES*16*128];  // 32 KB
  const int wave = threadIdx.x >> 5, lane = threadIdx.x & 31;
  float* st = stage + wave*(16*128);
  const int pixbase = blockIdx.x*(BWAVES*16) + wave*16;
  const int mA   = lane & 15;
  const int koff = (lane >> 4) * 2;
  const int mb   = (lane >> 4) * 8;

  v8f acc1[8];
#pragma unroll
  for (int t = 0; t < 8; ++t) acc1[t] = vzero8();

  // K = 1024 = 8 chunks; chunk kc == pooled-z slice kc (channel order zh*128 + c)
  for (int kc = 0; kc < 8; ++kc) {
    __syncthreads();
    {
      const int m = lane & 15;
      const int pix = pixbase + m;
      for (int j = 0; j < 64; ++j) {
        const int c = 2*j + (lane >> 4);
        const float* vp = vol_out + (size_t)c*L_TOT + (size_t)pix*16 + 2*kc;
        st[m*128 + c] = fmaxf(vp[0], vp[1]);     // MaxPool3d k=(1,1,2)
      }
    }
    __syncthreads();
    for (int t = 0; t < 8; ++t) {
      const v2f* bp = conv1_p + (size_t)t*8192 + (size_t)kc*1024 + lane;
      v8f a1 = acc1[t];
#pragma unroll 4
      for (int ks = 0; ks < 32; ++ks) {
        const int k0 = ks*4 + koff;
        v2f a; a.x = st[mA*128 + k0]; a.y = st[mA*128 + k0 + 1];
        a1 = wmma_f32(a, bp[ks*32], a1);
      }
      acc1[t] = a1;
    }
  }

  // h = relu(acc1 + conv1_b) -> st
  __syncthreads();
  for (int t = 0; t < 8; ++t) {
    const int col = t*16 + (lane & 15);
    const float bias = conv1_b[col];
#pragma unroll
    for (int r = 0; r < 8; ++r) st[(mb + r)*128 + col] = fmaxf(acc1[t][r] + bias, 0.f);
  }
  __syncthreads();

  // bev_f = h @ conv2_w + conv2_b
  v8f acc2[8];
  for (int t = 0; t < 8; ++t) {
    const v2f* bp = conv2_p + (size_t)t*1024 + lane;
    v8f a2 = vzero8();
#pragma unroll 4
    for (int ks = 0; ks < 32; ++ks) {
      const int k0 = ks*4 + koff;
      v2f a; a.x = st[mA*128 + k0]; a.y = st[mA*128 + k0 + 1];
      a2 = wmma_f32(a, bp[ks*32], a2);
    }
    acc2[t] = a2;
  }
  __syncthreads();
  for (int t = 0; t < 8; ++t) {
    const int col = t*16 + (lane & 15);
    const float bias = conv2_b[col];
#pragma unroll
    for (int r = 0; r < 8; ++r) {
      const float v = acc2[t][r] + bias;
      st[(mb + r)*128 + col] = v;
      bev_f_out[(size_t)col*(SX*SY) + pixbase + mb + r] = v;
    }
  }
  __syncthreads();

  // seg = bev_f @ seg_w + seg_b (N=20 -> 2 tiles, zero-padded in packed B)
  for (int t = 0; t < 2; ++t) {
    const int col = t*16 + (lane & 15);
    const v2f* bp = seg_p + (size_t)t*1024 + lane;
    v8f a3 = vzero8();
#pragma unroll 4
    for (int ks = 0; ks < 32; ++ks) {
      const int k0 = ks*4 + koff;
      v2f a; a.x = st[mA*128 + k0]; a.y = st[mA*128 + k0 + 1];
      a3 = wmma_f32(a, bp[ks*32], a3);
    }
    if (col < NUM_CLS) {
      const float bias = seg_b[col];
#pragma unroll
      for (int r = 0; r < 8; ++r)
        seg_out[(size_t)col*(SX*SY) + pixbase + mb + r] = a3[r] + bias;
    }
  }
}

// ---------------- host launcher ----------------
extern "C" void kernel_launch(void* const* d_in, const int* in_sizes, int n_in,
                              void* d_out, int out_size, void* d_ws, size_t ws_size,
                              hipStream_t stream) {
  (void)in_sizes; (void)n_in; (void)out_size; (void)ws_size;
  const float* volume      = (const float*)d_in[0];
  const float* feat0       = (const float*)d_in[1];
  const float* feat1       = (const float*)d_in[2];
  const float* feat2       = (const float*)d_in[3];
  const float* depth       = (const float*)d_in[4];
  const float* Kmat        = (const float*)d_in[5];
  const float* Emat        = (const float*)d_in[6];
  const float* vox_origin  = (const float*)d_in[7];
  const float* post_rot    = (const float*)d_in[8];
  const float* post_tran   = (const float*)d_in[9];
  const float* proj_pix    = (const float*)d_in[10];
  const float* pe_x        = (const float*)d_in[11];
  const float* pe_y        = (const float*)d_in[12];
  const float* pe_z        = (const float*)d_in[13];
  const float* pe_proj_w   = (const float*)d_in[14];
  const float* pe_proj_b   = (const float*)d_in[15];
  const float* w_off       = (const float*)d_in[16];
  const float* b_off       = (const float*)d_in[17];
  const float* w_attn      = (const float*)d_in[18];
  const float* b_attn      = (const float*)d_in[19];
  const float* w_val       = (const float*)d_in[20];
  const float* b_val       = (const float*)d_in[21];
  const float* w_out       = (const float*)d_in[22];
  const float* b_out       = (const float*)d_in[23];
  const float* ln_g        = (const float*)d_in[24];
  const float* ln_b        = (const float*)d_in[25];
  const float* conv1_w     = (const float*)d_in[26];
  const float* conv1_b     = (const float*)d_in[27];
  const float* conv2_w     = (const float*)d_in[28];
  const float* conv2_b     = (const float*)d_in[29];
  const float* seg_w       = (const float*)d_in[30];
  const float* seg_b       = (const float*)d_in[31];

  // workspace layout (floats; all offsets kept 8-byte aligned)
  float* ws       = (float*)d_ws;
  float* ws_inv   = ws;                         // 32
  float* px_proj  = ws + 32;                    // 128*512
  float* py_proj  = px_proj + 128*512;          // 128*512
  float* pz_proj  = py_proj + 128*512;          // 4*512
  float* vmap     = pz_proj + 4*512;            // 2520*128
  unsigned* mask  = (unsigned*)(vmap + VROWS*EMBED);  // L_TOT u32
  float* packs    = (float*)(mask + L_TOT);
  float* woff_p   = packs;                      // 12*32*64 = 24576
  float* wattn_p  = woff_p  + 12*32*64;         //  6*32*64 = 12288
  float* wout_p   = wattn_p +  6*32*64;         //  8*32*64 = 16384
  float* conv1_p  = wout_p  +  8*32*64;         //  8*256*64 = 131072
  float* conv2_p  = conv1_p +  8*256*64;        //  8*32*64 = 16384
  float* seg_p    = conv2_p +  8*32*64;         //  2*32*64 = 4096

  float* vol_out = (float*)d_out;                       // 128*L
  float* bev_f   = vol_out + (size_t)EMBED * L_TOT;     // 128*16384
  float* seg     = bev_f + (size_t)EMBED * SX * SY;     // 20*16384

  prep_kernel<<<1, 1, 0, stream>>>(Kmat, Emat, post_rot, ws_inv);

  posproj_kernel<<<(128*512 + 255)/256, 256, 0, stream>>>(pe_x, pe_proj_w, nullptr, px_proj, 128);
  posproj_kernel<<<(128*512 + 255)/256, 256, 0, stream>>>(pe_y, pe_proj_w, nullptr, py_proj, 128);
  posproj_kernel<<<(4*512 + 255)/256, 256, 0, stream>>>(pe_z, pe_proj_w, pe_proj_b, pz_proj, 4);

  valproj_kernel<<<(VROWS*EMBED + 255)/256, 256, 0, stream>>>(feat0, feat1, feat2, w_val, b_val, vmap);

  // pack all WMMA B operands into fragment order
  packB_kernel<<<(12*32*64 + 255)/256, 256, 0, stream>>>(w_off,   woff_p,  128, 192, 12);
  packB_kernel<<<( 6*32*64 + 255)/256, 256, 0, stream>>>(w_attn,  wattn_p, 128,  96,  6);
  packB_kernel<<<( 8*32*64 + 255)/256, 256, 0, stream>>>(w_out,   wout_p,  128, 128,  8);
  packB_kernel<<<(8*256*64 + 255)/256, 256, 0, stream>>>(conv1_w, conv1_p, 1024, 128, 8);
  packB_kernel<<<( 8*32*64 + 255)/256, 256, 0, stream>>>(conv2_w, conv2_p, 128, 128,  8);
  packB_kernel<<<( 2*32*64 + 255)/256, 256, 0, stream>>>(seg_w,   seg_p,   128, NUM_CLS, 2);

  maskzero_kernel<<<(L_TOT + 255)/256, 256, 0, stream>>>(mask);
  maskscatter_kernel<<<(NPIX + 255)/256, 256, 0, stream>>>(depth, vox_origin, post_tran, ws_inv, mask);

  attn_kernel<<<L_TOT/(CWAVES*16), CWAVES*32, 0, stream>>>(
      volume, proj_pix, px_proj, py_proj, pz_proj,
      (const v2f*)woff_p, (const v2f*)wattn_p, (const v2f*)wout_p,
      b_off, b_attn, vmap, b_out, ln_g, ln_b, mask, vol_out);

  bev_kernel<<<(SX*SY)/(BWAVES*16), BWAVES*32, 0, stream>>>(
      vol_out, (const v2f*)conv1_p, conv1_b, (const v2f*)conv2_p, conv2_b,
      (const v2f*)seg_p, seg_b, bev_f, seg);
}